// GNN_51857435132416
// MI455X (gfx1250) — compile-verified
//
#include <hip/hip_runtime.h>
#include <hip/hip_bf16.h>
#include <math.h>

#define N_NODES   50000
#define N_EDGES   2400000
#define NUM_HOPS  3
#define NUM_GRAPHS 2048
#define HIDDEN    128
#define EMBD      64
#define N_LAYERS  3
#define N_MLP     2
#define N_HEADS   4
#define FEATS     ((NUM_HOPS + 1) * HIDDEN)   // 512

typedef __attribute__((ext_vector_type(2))) float v2f;
typedef __attribute__((ext_vector_type(8))) float v8f;

__device__ __forceinline__ float silu_f(float v) { return v / (1.0f + expf(-v)); }

// order-preserving float<->uint encoding for atomic segment-max
__device__ __forceinline__ unsigned enc_f(float f) {
  unsigned b = __float_as_uint(f);
  return (b & 0x80000000u) ? ~b : (b | 0x80000000u);
}
__device__ __forceinline__ float dec_f(unsigned u) {
  return (u & 0x80000000u) ? __uint_as_float(u & 0x7FFFFFFFu) : __uint_as_float(~u);
}

// ---------------- embedding concat: emb = [e_atom|e_h|e_deg|e_hyb]  [N,256] ---
__global__ void embed_kernel(const int* __restrict__ atom_type, const int* __restrict__ hyd,
                             const int* __restrict__ deg, const int* __restrict__ hyb,
                             const float* __restrict__ e_atom, const float* __restrict__ e_h,
                             const float* __restrict__ e_deg, const float* __restrict__ e_hyb,
                             float* __restrict__ emb) {
  int i = blockIdx.x * blockDim.x + threadIdx.x;
  if (i >= N_NODES * 4 * EMBD) return;
  int n = i >> 8;         // /256
  int f = i & 255;
  int seg = f >> 6, j = f & 63;
  float v;
  if      (seg == 0) v = e_atom[atom_type[n] * EMBD + j];
  else if (seg == 1) v = e_h[hyd[n] * EMBD + j];
  else if (seg == 2) v = e_deg[deg[n] * EMBD + j];
  else               v = e_hyb[hyb[n] * EMBD + j];
  emb[i] = v;
}

// ---------------- f32 WMMA GEMM: out[M,128] = op(A[M,K] @ W[K,128] + b) --------
// Block = 256 threads = 8 waves covering a 64x128 output tile.
// Each wave owns a 32x32 tile via 4 accumulators: per K-step (K+=4) it loads
// 2 A fragments + 2 B fragments and issues 4 v_wmma_f32_16x16x4_f32 -> 2x
// operand reuse on A and B vs a 16x16 tile.
// act: 1 -> SiLU.  res1/res2: optional elementwise residual adds (post-act).
__global__ void __launch_bounds__(256)
wmma_gemm(const float* __restrict__ A, const float* __restrict__ W,
          const float* __restrict__ bias, const float* __restrict__ res1,
          const float* __restrict__ res2, float* __restrict__ out,
          int M, int K, int act) {
  const int wave = threadIdx.x >> 5;
  const int lane = threadIdx.x & 31;
  const int hl   = lane >> 4;          // 0: lanes 0-15 (K 0,1) | 1: lanes 16-31 (K 2,3)
  const int idx  = lane & 15;
  const int row0 = blockIdx.x * 64 + (wave >> 2) * 32;   // 2 row-groups of waves
  const int col0 = (wave & 3) * 32;                      // 4 col-groups of waves

  v8f c00 = {0.f,0.f,0.f,0.f,0.f,0.f,0.f,0.f};
  v8f c01 = c00, c10 = c00, c11 = c00;

  const int   ar0 = row0 + idx,            ar1 = row0 + 16 + idx;
  const int   arc0 = (ar0 < M) ? ar0 : 0,  arc1 = (ar1 < M) ? ar1 : 0;
  const float m0 = (ar0 < M) ? 1.0f : 0.0f;
  const float m1 = (ar1 < M) ? 1.0f : 0.0f;
  const float* __restrict__ Ap0 = A + (size_t)arc0 * K + 2 * hl;
  const float* __restrict__ Ap1 = A + (size_t)arc1 * K + 2 * hl;
  const float* __restrict__ Wp0 = W + (size_t)(2 * hl) * HIDDEN + col0 + idx;
  const float* __restrict__ Wp1 = Wp0 + 16;

  #pragma unroll 4
  for (int k = 0; k < K; k += 4) {
    __builtin_prefetch(Ap0 + k + 64, 0, 1);   // global_prefetch of the A stream
    __builtin_prefetch(Ap1 + k + 64, 0, 1);
    v2f a0, a1, b0, b1;
    a0.x = m0 * Ap0[k];  a0.y = m0 * Ap0[k + 1];
    a1.x = m1 * Ap1[k];  a1.y = m1 * Ap1[k + 1];
    b0.x = Wp0[(size_t)k * HIDDEN];  b0.y = Wp0[(size_t)(k + 1) * HIDDEN];
    b1.x = Wp1[(size_t)k * HIDDEN];  b1.y = Wp1[(size_t)(k + 1) * HIDDEN];
    c00 = __builtin_amdgcn_wmma_f32_16x16x4_f32(false, a0, false, b0, (short)0, c00, false, false);
    c01 = __builtin_amdgcn_wmma_f32_16x16x4_f32(false, a0, false, b1, (short)0, c01, false, false);
    c10 = __builtin_amdgcn_wmma_f32_16x16x4_f32(false, a1, false, b0, (short)0, c10, false, false);
    c11 = __builtin_amdgcn_wmma_f32_16x16x4_f32(false, a1, false, b1, (short)0, c11, false, false);
  }

  auto store_tile = [&](v8f c, int tr0, int tc0) {
    const float bcol = bias[tc0 + idx];
    #pragma unroll
    for (int r = 0; r < 8; ++r) {
      int row = tr0 + r + 8 * hl;      // C layout: VGPR r -> M=r (lanes 0-15) / M=8+r
      if (row < M) {
        float v = c[r] + bcol;
        if (act) v = silu_f(v);
        size_t o = (size_t)row * HIDDEN + tc0 + idx;
        if (res1) v += res1[o];
        if (res2) v += res2[o];
        out[o] = v;
      }
    }
  };
  store_tile(c00, row0,      col0);
  store_tile(c01, row0,      col0 + 16);
  store_tile(c10, row0 + 16, col0);
  store_tile(c11, row0 + 16, col0 + 16);
}

// ---------------- x -> F[:, 0:128] (F row stride = 512) ----------------
__global__ void copy_x_to_F(const float* __restrict__ x, float* __restrict__ F) {
  int i = blockIdx.x * blockDim.x + threadIdx.x;
  if (i >= N_NODES * HIDDEN) return;
  int n = i >> 7, f = i & 127;
  F[(size_t)n * FEATS + f] = x[i];
}

// ---------------- edge scatter-add into F[:, 128*(hop+1) ...] ----------------
// agg[target] += x[src % N]; target = hop*N + node
__global__ void __launch_bounds__(256)
edge_scatter(const int* __restrict__ tgt, const int* __restrict__ src,
             const float* __restrict__ x, float* __restrict__ F) {
  int e = blockIdx.x * 8 + (threadIdx.x >> 5);
  if (e >= N_EDGES) return;
  int lane = threadIdx.x & 31;
  int s  = src[e] % N_NODES;
  int t  = tgt[e];
  int th = t / N_NODES;
  int tn = t - th * N_NODES;
  const float4 v = *reinterpret_cast<const float4*>(x + (size_t)s * HIDDEN + lane * 4);
  float* dst = F + (size_t)tn * FEATS + (th + 1) * HIDDEN + lane * 4;
  atomicAdd(dst + 0, v.x);
  atomicAdd(dst + 1, v.y);
  atomicAdd(dst + 2, v.z);
  atomicAdd(dst + 3, v.w);
}

// ---------------- attention: scores + segment max (1 wave per node) ----------
__global__ void __launch_bounds__(256)
attn_scores(const float* __restrict__ x, const float* __restrict__ aw,
            const float* __restrict__ ab, const float* __restrict__ temp,
            const int* __restrict__ batch, float* __restrict__ scores,
            unsigned* __restrict__ smax) {
  int n = blockIdx.x * 8 + (threadIdx.x >> 5);
  if (n >= N_NODES) return;
  int lane = threadIdx.x & 31;
  float invT = 1.0f / temp[0];
  int g = batch[n];
  const float4 xv = *reinterpret_cast<const float4*>(x + (size_t)n * HIDDEN + lane * 4);
  #pragma unroll
  for (int h = 0; h < N_HEADS; ++h) {
    const float4 wv = *reinterpret_cast<const float4*>(aw + h * HIDDEN + lane * 4);
    float p = xv.x * wv.x + xv.y * wv.y + xv.z * wv.z + xv.w * wv.w;
    #pragma unroll
    for (int m = 16; m > 0; m >>= 1) p += __shfl_xor(p, m, 32);
    if (lane == 0) {
      float s = (p + ab[h]) * invT;
      scores[n * N_HEADS + h] = s;
      atomicMax(&smax[g * N_HEADS + h], enc_f(s));
    }
  }
}

// ---------------- exp(scores - max) + segment sum ----------------
__global__ void attn_exp(const float* __restrict__ scores, const unsigned* __restrict__ smax,
                         const int* __restrict__ batch, float* __restrict__ ex,
                         float* __restrict__ denom) {
  int n = blockIdx.x * blockDim.x + threadIdx.x;
  if (n >= N_NODES) return;
  int g = batch[n];
  #pragma unroll
  for (int h = 0; h < N_HEADS; ++h) {
    float e = expf(scores[n * N_HEADS + h] - dec_f(smax[g * N_HEADS + h]));
    ex[n * N_HEADS + h] = e;
    atomicAdd(&denom[g * N_HEADS + h], e);
  }
}

// ---------------- weighted segment pooling: pooled[g, h*128+f] ----------------
__global__ void __launch_bounds__(128)
attn_pool(const float* __restrict__ x, const float* __restrict__ ex,
          const float* __restrict__ denom, const int* __restrict__ batch,
          float* __restrict__ pooled) {
  int n = blockIdx.x;
  int f = threadIdx.x;
  int g = batch[n];
  float xv = x[(size_t)n * HIDDEN + f];
  #pragma unroll
  for (int h = 0; h < N_HEADS; ++h) {
    float w = ex[n * N_HEADS + h] / denom[g * N_HEADS + h];
    atomicAdd(&pooled[(size_t)g * (N_HEADS * HIDDEN) + h * HIDDEN + f], xv * w);
  }
}

// ---------------- mean over heads ----------------
__global__ void head_mean(const float* __restrict__ pooled, float* __restrict__ pm) {
  int i = blockIdx.x * blockDim.x + threadIdx.x;
  if (i >= NUM_GRAPHS * HIDDEN) return;
  int g = i >> 7, f = i & 127;
  float s = 0.f;
  #pragma unroll
  for (int h = 0; h < N_HEADS; ++h) s += pooled[(size_t)g * (N_HEADS * HIDDEN) + h * HIDDEN + f];
  pm[i] = 0.25f * s;
}

// ---------------- final 128 -> 1 projection ----------------
__global__ void ffn_out(const float* __restrict__ h, const float* __restrict__ w3,
                        const float* __restrict__ b3, float* __restrict__ out) {
  int g = blockIdx.x * blockDim.x + threadIdx.x;
  if (g >= NUM_GRAPHS) return;
  float s = b3[0];
  for (int f = 0; f < HIDDEN; ++f) s += h[(size_t)g * HIDDEN + f] * w3[f];
  out[g] = s;
}

static inline int ceil_div(long long a, long long b) { return (int)((a + b - 1) / b); }

extern "C" void kernel_launch(void* const* d_in, const int* in_sizes, int n_in,
                              void* d_out, int out_size, void* d_ws, size_t ws_size,
                              hipStream_t stream) {
  const int*   atom_type = (const int*)d_in[0];
  const int*   hyd       = (const int*)d_in[1];
  const int*   deg       = (const int*)d_in[2];
  const int*   hyb       = (const int*)d_in[3];
  const int*   target    = (const int*)d_in[4];
  const int*   src       = (const int*)d_in[5];
  const int*   batch     = (const int*)d_in[6];
  const float* e_atom    = (const float*)d_in[7];
  const float* e_h       = (const float*)d_in[8];
  const float* e_deg     = (const float*)d_in[9];
  const float* e_hyb     = (const float*)d_in[10];
  const float* proj_w    = (const float*)d_in[11];
  const float* proj_b    = (const float*)d_in[12];
  const float* sin_w     = (const float*)d_in[13];
  const float* sin_b     = (const float*)d_in[14];
  const float* mlp_w1    = (const float*)d_in[15];
  const float* mlp_b1    = (const float*)d_in[16];
  const float* mlp_w2    = (const float*)d_in[17];
  const float* mlp_b2    = (const float*)d_in[18];
  const float* skp_w     = (const float*)d_in[19];
  const float* skp_b     = (const float*)d_in[20];
  const float* attn_w    = (const float*)d_in[21];
  const float* attn_b    = (const float*)d_in[22];
  const float* temp      = (const float*)d_in[23];
  const float* ffn_w1    = (const float*)d_in[24];
  const float* ffn_b1    = (const float*)d_in[25];
  const float* ffn_w2    = (const float*)d_in[26];
  const float* ffn_b2    = (const float*)d_in[27];
  const float* ffn_w3    = (const float*)d_in[28];
  const float* ffn_b3    = (const float*)d_in[29];

  // ---- workspace partition ----
  char* ws = (char*)d_ws;
  size_t off = 0;
  auto alloc = [&](size_t bytes) -> void* {
    void* p = ws + off;
    off = (off + bytes + 255) & ~(size_t)255;
    return p;
  };
  float*    emb    = (float*)alloc((size_t)N_NODES * 4 * EMBD * 4);
  float*    x      = (float*)alloc((size_t)N_NODES * HIDDEN * 4);
  float*    F      = (float*)alloc((size_t)N_NODES * FEATS * 4);
  float*    bufH   = (float*)alloc((size_t)N_NODES * HIDDEN * 4);
  float*    bufT   = (float*)alloc((size_t)N_NODES * HIDDEN * 4);
  float*    bufG   = (float*)alloc((size_t)N_NODES * HIDDEN * 4);
  float*    scores = (float*)alloc((size_t)N_NODES * N_HEADS * 4);
  float*    exb    = (float*)alloc((size_t)N_NODES * N_HEADS * 4);
  unsigned* smax   = (unsigned*)alloc((size_t)NUM_GRAPHS * N_HEADS * 4);
  float*    denom  = (float*)alloc((size_t)NUM_GRAPHS * N_HEADS * 4);
  float*    pooled = (float*)alloc((size_t)NUM_GRAPHS * N_HEADS * HIDDEN * 4);
  float*    pm     = (float*)alloc((size_t)NUM_GRAPHS * HIDDEN * 4);
  float*    f1     = (float*)alloc((size_t)NUM_GRAPHS * HIDDEN * 4);
  float*    f2     = (float*)alloc((size_t)NUM_GRAPHS * HIDDEN * 4);

  const int gemm_blocks_N = ceil_div(N_NODES, 64);
  const int gemm_blocks_G = ceil_div(NUM_GRAPHS, 64);

  // 1) embedding + projection: x = silu(emb @ proj_w + b)
  embed_kernel<<<ceil_div((long long)N_NODES * 4 * EMBD, 256), 256, 0, stream>>>(
      atom_type, hyd, deg, hyb, e_atom, e_h, e_deg, e_hyb, emb);
  wmma_gemm<<<gemm_blocks_N, 256, 0, stream>>>(emb, proj_w, proj_b, nullptr, nullptr,
                                               x, N_NODES, 4 * EMBD, 1);

  // 2) shell conv layers
  for (int l = 0; l < N_LAYERS; ++l) {
    hipMemsetAsync(F, 0, (size_t)N_NODES * FEATS * 4, stream);
    copy_x_to_F<<<ceil_div((long long)N_NODES * HIDDEN, 256), 256, 0, stream>>>(x, F);
    edge_scatter<<<ceil_div(N_EDGES, 8), 256, 0, stream>>>(target, src, x, F);

    wmma_gemm<<<gemm_blocks_N, 256, 0, stream>>>(
        F, sin_w + (size_t)l * FEATS * HIDDEN, sin_b + l * HIDDEN,
        nullptr, nullptr, bufH, N_NODES, FEATS, 1);
    wmma_gemm<<<gemm_blocks_N, 256, 0, stream>>>(
        F, skp_w + (size_t)l * FEATS * HIDDEN, skp_b + l * HIDDEN,
        nullptr, nullptr, bufG, N_NODES, FEATS, 0);

    for (int m = 0; m < N_MLP; ++m) {
      const size_t wo = (size_t)(l * N_MLP + m) * HIDDEN * HIDDEN;
      const size_t bo = (size_t)(l * N_MLP + m) * HIDDEN;
      wmma_gemm<<<gemm_blocks_N, 256, 0, stream>>>(
          bufH, mlp_w1 + wo, mlp_b1 + bo, nullptr, nullptr, bufT, N_NODES, HIDDEN, 1);
      if (m == N_MLP - 1) {
        // x = (t @ w2 + b2) + h + gskip   (fused epilogue)
        wmma_gemm<<<gemm_blocks_N, 256, 0, stream>>>(
            bufT, mlp_w2 + wo, mlp_b2 + bo, bufH, bufG, x, N_NODES, HIDDEN, 0);
      } else {
        // h = (t @ w2 + b2) + h   (element-wise in-place residual is safe)
        wmma_gemm<<<gemm_blocks_N, 256, 0, stream>>>(
            bufT, mlp_w2 + wo, mlp_b2 + bo, bufH, nullptr, bufH, N_NODES, HIDDEN, 0);
      }
    }
  }

  // 3) attention pooling with segment softmax
  hipMemsetAsync(smax, 0, (size_t)NUM_GRAPHS * N_HEADS * 4, stream);   // 0 < enc(any float)
  hipMemsetAsync(denom, 0, (size_t)NUM_GRAPHS * N_HEADS * 4, stream);
  hipMemsetAsync(pooled, 0, (size_t)NUM_GRAPHS * N_HEADS * HIDDEN * 4, stream);
  attn_scores<<<ceil_div(N_NODES, 8), 256, 0, stream>>>(x, attn_w, attn_b, temp, batch,
                                                        scores, smax);
  attn_exp<<<ceil_div(N_NODES, 256), 256, 0, stream>>>(scores, smax, batch, exb, denom);
  attn_pool<<<N_NODES, 128, 0, stream>>>(x, exb, denom, batch, pooled);
  head_mean<<<ceil_div((long long)NUM_GRAPHS * HIDDEN, 256), 256, 0, stream>>>(pooled, pm);

  // 4) FFN readout
  wmma_gemm<<<gemm_blocks_G, 256, 0, stream>>>(pm, ffn_w1, ffn_b1, nullptr, nullptr,
                                               f1, NUM_GRAPHS, HIDDEN, 1);
  wmma_gemm<<<gemm_blocks_G, 256, 0, stream>>>(f1, ffn_w2, ffn_b2, nullptr, nullptr,
                                               f2, NUM_GRAPHS, HIDDEN, 1);
  ffn_out<<<ceil_div(NUM_GRAPHS, 256), 256, 0, stream>>>(f2, ffn_w3, ffn_b3, (float*)d_out);
}